// controller_60387240181829
// MI455X (gfx1250) — compile-verified
//
#include <hip/hip_runtime.h>
#include <hip/hip_bf16.h>

// Sequential chain of 320 3-layer LSTM cells (H=3, G=12); carry (h,c) flows
// cell->cell, so the kernel is pure dependent-chain latency. One compute wave
// runs the chain; 7 helper waves double-buffer weights into LDS one chunk
// ahead (GLOBAL_LOAD_ASYNC_TO_LDS_B128, ASYNCcnt-drained before the barrier).
// Gates: two chained V_WMMA_F32_16X16X4_F32 with A = broadcast vector and
// B = [W | bias]^T, so the 12 gates land lane-striped in D VGPR0.
// Activations: one per-lane-parameterized a + b*rcp(1+exp2(c*x)) chain.
// hreg lane 15 is pinned to 1.0 so A-operand builds are pure ds_bpermute
// (no exec-masked shuffles anywhere on the critical path).

#define N_CELLS 320
#define CHUNK   16
#define NCH     (N_CELLS / CHUNK)
#define NT_COPY 224

// Flat LDS layout (float elements) per buffer:
#define OFF_WHH   0        // CHUNK*108 : [ci][l][n][k]
#define OFF_WIH12 1728     // CHUNK*72  : [ci][l-1][n][k]
#define OFF_WIH0  2880     // CHUNK*12  : [ci][n]
#define OFF_BIH   3072     // CHUNK*36  : [ci][l][n]
#define OFF_BHH   3648     // CHUNK*36  : [ci][l][n]
#define BUFSZ     4224

#define LOG2E 1.4426950408889634f

typedef __attribute__((ext_vector_type(2))) float v2f;
typedef __attribute__((ext_vector_type(8))) float v8f;

#if __has_builtin(__builtin_amdgcn_global_load_async_to_lds_b128)
#define HAVE_ASYNC 1
typedef int v4i_t __attribute__((ext_vector_type(4)));
typedef __attribute__((address_space(1))) v4i_t* gv4i_p;  // global int4*
typedef __attribute__((address_space(3))) v4i_t* lv4i_p;  // LDS int4*
#endif

__device__ __forceinline__ float fexp2(float x) { return __builtin_amdgcn_exp2f(x); } // v_exp_f32
__device__ __forceinline__ float frcp(float x)  { return __builtin_amdgcn_rcpf(x); }  // v_rcp_f32
__device__ __forceinline__ float ftanh(float x) { return 1.0f - 2.0f * frcp(1.0f + fexp2(2.0f * LOG2E * x)); }

__device__ __forceinline__ void copy_region(const float* __restrict__ g, float* l,
                                            int nf4, int j, int nt) {
#ifdef HAVE_ASYNC
    for (int k = j; k < nf4; k += nt) {
        __builtin_amdgcn_global_load_async_to_lds_b128(
            (gv4i_p)(float*)(g + 4 * k),
            (lv4i_p)(l + 4 * k), 0, 0);
    }
#else
    const float4* s = (const float4*)g;
    float4* d = (float4*)l;
    for (int k = j; k < nf4; k += nt) d[k] = s[k];
#endif
}

__device__ __forceinline__ void stage_chunk(int t, float* buf,
    const float* __restrict__ w_hh,  const float* __restrict__ w_ih12,
    const float* __restrict__ w_ih0, const float* __restrict__ b_ih,
    const float* __restrict__ b_hh,  int j, int nt)
{
    copy_region(w_hh   + (size_t)t * (CHUNK * 108), buf + OFF_WHH,   CHUNK * 108 / 4, j, nt);
    copy_region(w_ih12 + (size_t)t * (CHUNK * 72),  buf + OFF_WIH12, CHUNK * 72  / 4, j, nt);
    copy_region(w_ih0  + (size_t)t * (CHUNK * 12),  buf + OFF_WIH0,  CHUNK * 12  / 4, j, nt);
    copy_region(b_ih   + (size_t)t * (CHUNK * 36),  buf + OFF_BIH,   CHUNK * 36  / 4, j, nt);
    copy_region(b_hh   + (size_t)t * (CHUNK * 36),  buf + OFF_BHH,   CHUNK * 36  / 4, j, nt);
#ifdef HAVE_ASYNC
    asm volatile("s_wait_asynccnt 0x0" ::: "memory");  // drain before barrier
#endif
}

__global__ __launch_bounds__(256, 1)
void lstm_chain_wmma_kernel(const float* __restrict__ state,
                            const float* __restrict__ w_ih0,
                            const float* __restrict__ w_ih12,
                            const float* __restrict__ w_hh,
                            const float* __restrict__ b_ih,
                            const float* __restrict__ b_hh,
                            float* __restrict__ out)
{
    __shared__ __align__(16) float lds[2 * BUFSZ + N_CELLS];
    float* sstate = lds + 2 * BUFSZ;

    const int tid  = threadIdx.x;
    const int lane = tid & 31;
    const int wv   = tid >> 5;

    for (int k = tid; k < N_CELLS; k += 256) sstate[k] = state[k];
    stage_chunk(0, lds, w_hh, w_ih12, w_ih0, b_ih, b_hh, tid, 256);
    __syncthreads();

    // Carry: lane (3*l + u) holds h[l][u] / c[l][u]. Lane 15 of hreg is
    // pinned to 1.0 (the bias multiplier) -- updates only touch lanes 0..8.
    float hreg = (lane == 15) ? 1.0f : 0.0f;
    float creg = 0.0f;

    // Per-lane activation constants (hoisted): act = ka + kb*rcp(1+exp2(kc*x))
    //   gate lanes 0-5, 9-11 (i,f,o): sigmoid -> ka=0, kb=1,  kc=-log2e
    //   gate lanes 6-8        (g)   : tanh    -> ka=1, kb=-2, kc=2*log2e
    const bool  isg = ((unsigned)(lane - 6) < 3u);
    const float ka  = isg ? 1.0f : 0.0f;
    const float kb  = isg ? -2.0f : 1.0f;
    const float kc  = isg ? (2.0f * LOG2E) : (-LOG2E);

    for (int t = 0; t < NCH; ++t) {
        if (wv != 0) {
            if (t + 1 < NCH)
                stage_chunk(t + 1, lds + ((t + 1) & 1) * BUFSZ,
                            w_hh, w_ih12, w_ih0, b_ih, b_hh, tid - 32, NT_COPY);
            if (t + 2 < NCH) {  // warm L2 two chunks ahead (global_prefetch_b8)
                const char* p = (const char*)(w_hh + (size_t)(t + 2) * (CHUNK * 108));
                __builtin_prefetch(p + (size_t)(tid - 32) * 128, 0, 1);
                const char* q = (const char*)(w_ih12 + (size_t)(t + 2) * (CHUNK * 72));
                __builtin_prefetch(q + (size_t)(tid - 32) * 64, 0, 1);
            }
        } else {
            const int base = (t & 1) * BUFSZ;
            const int hi   = (lane >= 16);
            const int k02  = hi ? 2 : 0;       // K-row this half-lane holds in VGPR0
            int n = lane & 15;
            if (n >= 12) n = 0;                // cols 12..15 never read
            const int n3 = n * 3;

            for (int ci = 0; ci < CHUNK; ++ci) {
                const int cell = t * CHUNK + ci;
                const float x = sstate[cell];

#pragma unroll
                for (int l = 0; l < 3; ++l) {
                    // A operands: v_h = [h_l0,h_l1,h_l2,1], v_x = [x0,x1,x2,1].
                    // K3's 1.0 comes from hreg lane 15 -> plain ds_bpermute,
                    // no exec-masked shuffle, indices loop-invariant per lane.
                    v2f ah, ax, b1, b2;
                    ah.x = __shfl(hreg, 3 * l + k02, 32);
                    ah.y = __shfl(hreg, hi ? 15 : (3 * l + 1), 32);
                    if (l == 0) {
                        ax.x = hi ? 0.0f : x;          // x is wave-uniform
                        ax.y = hi ? 1.0f : 0.0f;
                    } else {
                        ax.x = __shfl(hreg, 3 * (l - 1) + k02, 32);
                        ax.y = __shfl(hreg, hi ? 15 : (3 * (l - 1) + 1), 32);
                    }

                    // Branchless per-lane LDS addressing (single ds_load each).
                    const int o_whh = base + OFF_WHH + ci * 108 + l * 36 + n3;
                    const int o_bih = base + OFF_BIH + ci * 36 + l * 12 + n;
                    const int o_bhh = base + OFF_BHH + ci * 36 + l * 12 + n;
                    b1.x = lds[o_whh + k02];
                    b1.y = lds[hi ? o_bih : (o_whh + 1)];
                    if (l == 0) {
                        float w0 = lds[base + OFF_WIH0 + ci * 12 + n];
                        float bh = lds[o_bhh];
                        b2.x = hi ? 0.0f : w0;
                        b2.y = hi ? bh : 0.0f;
                    } else {
                        const int o_wih = base + OFF_WIH12 + ci * 72 + (l - 1) * 36 + n3;
                        b2.x = lds[o_wih + k02];
                        b2.y = lds[hi ? o_bhh : (o_wih + 1)];
                    }

                    // gates = A_h*[Whh|b_ih]^T + A_x*[Wih|b_hh]^T
                    v8f acc = {0.f, 0.f, 0.f, 0.f, 0.f, 0.f, 0.f, 0.f};
                    acc = __builtin_amdgcn_wmma_f32_16x16x4_f32(
                        false, ah, false, b1, (short)0, acc, false, false);
                    acc = __builtin_amdgcn_wmma_f32_16x16x4_f32(
                        false, ax, false, b2, (short)0, acc, false, false);
                    float gx = acc[0];   // lane holds gate (lane&15)

                    // Unified branch-free activation (sigmoid or tanh by lane).
                    float act = ka + kb * frcp(1.0f + fexp2(kc * gx));

                    // Gather i/f/g/o for unit u into storage lane 3*l+u.
                    int ul = (lane - 3 * l) & 31;
                    float iv = __shfl(act, ul,           32);
                    float fv = __shfl(act, (ul + 3) & 31, 32);
                    float gv = __shfl(act, (ul + 6) & 31, 32);
                    float ov = __shfl(act, (ul + 9) & 31, 32);

                    float cn = fv * creg + iv * gv;
                    float hn = ov * ftanh(cn);
                    bool inr = ((unsigned)(lane - 3 * l) < 3u);   // lanes 3l..3l+2 only
                    hreg = inr ? hn : hreg;   // v_cndmask; lane 15 stays 1.0
                    creg = inr ? cn : creg;
                }

                // softmax over top-layer h (lanes 6..8); lanes 0..2 store.
                float t0 = __shfl(hreg, 6, 32);
                float t1 = __shfl(hreg, 7, 32);
                float t2 = __shfl(hreg, 8, 32);
                float m  = fmaxf(t0, fmaxf(t1, t2));
                float e0 = fexp2((t0 - m) * LOG2E);
                float e1 = fexp2((t1 - m) * LOG2E);
                float e2 = fexp2((t2 - m) * LOG2E);
                float rs = frcp(e0 + e1 + e2);
                if (lane < 3)
                    out[cell * 3 + lane] =
                        ((lane == 0) ? e0 : (lane == 1) ? e1 : e2) * rs;
            }
        }
        __syncthreads();
    }
}

extern "C" void kernel_launch(void* const* d_in, const int* in_sizes, int n_in,
                              void* d_out, int out_size, void* d_ws, size_t ws_size,
                              hipStream_t stream) {
    (void)in_sizes; (void)n_in; (void)d_ws; (void)ws_size; (void)out_size;
    const float* state  = (const float*)d_in[0];
    const float* w_ih0  = (const float*)d_in[1];
    const float* w_ih12 = (const float*)d_in[2];
    const float* w_hh   = (const float*)d_in[3];
    const float* b_ih   = (const float*)d_in[4];
    const float* b_hh   = (const float*)d_in[5];
    float* out = (float*)d_out;

    lstm_chain_wmma_kernel<<<1, 256, 0, stream>>>(
        state, w_ih0, w_ih12, w_hh, b_ih, b_hh, out);
}